// GlassBlur_22213570854917
// MI455X (gfx1250) — compile-verified
//
#include <hip/hip_runtime.h>
#include <cmath>

// ---------------------------------------------------------------------------
// GlassBlur for MI455X (gfx1250): swap via packed u64 atomicMax, separable
// 17-tap gaussian as banded-matrix WMMA (V_WMMA_F32_16X16X4_F32 chains).
// Band weights computed arithmetically (v_exp_f32) -> no predicated loads,
// EXEC stays all-1s through the WMMA chain.
// ---------------------------------------------------------------------------

typedef __attribute__((ext_vector_type(2))) float v2f;
typedef __attribute__((ext_vector_type(8))) float v8f;

#define IMG_H 2048
#define IMG_W 2048
#define NCH   3
#define HV    2045          // H - 2*KSIZE + 1
#define NPIX  (IMG_H * IMG_W)          // 4,194,304 = 2^22
#define NLOC  (HV * HV)                // 4,182,025
#define SRAD2 589824       // 768^2
#define SCLN2 586756       // (768-2)^2
#define RADIUS 8           // int(4*sigma+0.5), sigma=2

// Band weight: w(d) = exp(-0.5*((d-8)/2)^2)/S for d in [0,16], else 0.
// invS passed from host (sum over the same 17 float32 exp values).
__device__ __forceinline__ float bandw(int d, float invS) {
    float t = (float)(d - 8);
    float w = __expf(-0.125f * t * t) * invS;
    return ((unsigned)d <= 16u) ? w : 0.0f;
}

// ---------------- stage 0: init win keys ----------------
__global__ __launch_bounds__(256) void gb_init(unsigned long long* __restrict__ win) {
    int p = blockIdx.x * blockDim.x + threadIdx.x;
    if (p < NPIX) win[p] = 0ull;
}

// ---------------- stage 1: scatter-max of (order,src) keys ----------------
__global__ __launch_bounds__(256) void gb_scatter(const int* __restrict__ off,
                                                  unsigned long long* __restrict__ win) {
    int idx = blockIdx.x * blockDim.x + threadIdx.x;
    if (idx >= NLOC) return;
    int a = idx / HV;            // local row
    int b = idx - a * HV;        // local col
    int y = a + 2;               // HH
    int x = b + 2;               // WW
    int dx = 1024 - x, dy = 1024 - y;
    int d2 = dx * dx + dy * dy;
    if (d2 < SRAD2) return;      // swap_mask false -> all writes are -1
    int local = a * HV + b;
    unsigned lin = (unsigned)(y * IMG_W + x);
#pragma unroll
    for (int i = 0; i < 2; ++i) {
        int dh = off[((i * HV + a) * HV + b) * 2 + 0];
        int dw = off[((i * HV + a) * HV + b) * 2 + 1];
        unsigned lin2 = (unsigned)((y + dh) * IMG_W + (x + dw));
        unsigned long long o1 = (unsigned long long)(i * 2 * NLOC + 2 * local);
        // write: tgt=lin,  src=lin2, order=o1
        atomicMax(&win[lin],  ((o1 << 22) | lin2) + 1ull);
        // write: tgt=lin2, src=lin,  order=o1+1
        atomicMax(&win[lin2], (((o1 + 1ull) << 22) | lin) + 1ull);
    }
}

// ---------------- stage 2: gather winners + normalize to [0,1] -------------
__global__ __launch_bounds__(256) void gb_gather(const float* __restrict__ img,
                                                 const unsigned long long* __restrict__ win,
                                                 float* __restrict__ sw) {
    int p = blockIdx.x * blockDim.x + threadIdx.x;
    if (p >= NPIX) return;
    unsigned long long w = win[p];
    int s = (w != 0ull) ? (int)((w - 1ull) & 0x3FFFFFull) : p;
    const float inv = 1.0f / 255.0f;
    size_t sp = (size_t)s * NCH, dp = (size_t)p * NCH;
    sw[dp + 0] = img[sp + 0] * inv;
    sw[dp + 1] = img[sp + 1] * inv;
    sw[dp + 2] = img[sp + 2] * inv;
}

// ---------------- stage 3: horizontal blur via WMMA ------------------------
// One wave per 16x16 tile per channel. D = A(16x32 image rows) x B(32x16 band)
// accumulated as 8 chained V_WMMA_F32_16X16X4_F32.
__global__ __launch_bounds__(256) void gb_blur_h(const float* __restrict__ in,
                                                 float* __restrict__ out, float invS) {
    int wave = (blockIdx.x * blockDim.x + threadIdx.x) >> 5;
    int lane = threadIdx.x & 31;
    int c  = wave % NCH;
    int t  = wave / NCH;
    int tx = t & 127;
    int ty = t >> 7;
    int x0 = tx << 4, y0 = ty << 4;
    int N    = lane & 15;        // also M for the A operand
    int half = lane >> 4;
    int yA = y0 + N;             // A row held by this lane
    const float* rowp = in + ((size_t)yA * IMG_W) * NCH + c;
    v8f acc = {0.f, 0.f, 0.f, 0.f, 0.f, 0.f, 0.f, 0.f};
#pragma unroll
    for (int ch = 0; ch < 8; ++ch) {
        int j = 4 * ch + 2 * half;            // K slot 0 column
        int xa = x0 - RADIUS + j;
        int x1 = min(max(xa,     0), IMG_W - 1);   // edge padding
        int x2 = min(max(xa + 1, 0), IMG_W - 1);
        v2f a; a.x = rowp[(size_t)x1 * NCH]; a.y = rowp[(size_t)x2 * NCH];
        v2f b;                                 // B[j][N] = k[j-N]
        b.x = bandw(j - N,     invS);
        b.y = bandw(j + 1 - N, invS);
        acc = __builtin_amdgcn_wmma_f32_16x16x4_f32(false, a, false, b,
                                                    (short)0, acc, false, false);
    }
#pragma unroll
    for (int g = 0; g < 8; ++g) {
        int ry = y0 + g + 8 * half;
        out[((size_t)ry * IMG_W + (x0 + N)) * NCH + c] = acc[g];
    }
}

// ---------------- stage 4: vertical blur + clip*255 + clean blend ----------
// D = A(16x32 band) x B(32x16 image cols).
__global__ __launch_bounds__(256) void gb_blur_v(const float* __restrict__ in,
                                                 const float* __restrict__ img,
                                                 float* __restrict__ out, float invS) {
    int wave = (blockIdx.x * blockDim.x + threadIdx.x) >> 5;
    int lane = threadIdx.x & 31;
    int c  = wave % NCH;
    int t  = wave / NCH;
    int tx = t & 127;
    int ty = t >> 7;
    int x0 = tx << 4, y0 = ty << 4;
    int N    = lane & 15;        // A: M row of band; B: column offset
    int half = lane >> 4;
    v8f acc = {0.f, 0.f, 0.f, 0.f, 0.f, 0.f, 0.f, 0.f};
#pragma unroll
    for (int ch = 0; ch < 8; ++ch) {
        int j = 4 * ch + 2 * half;            // K slot 0 row
        v2f a;                                 // band[M=N][j] = k[j-N]
        a.x = bandw(j - N,     invS);
        a.y = bandw(j + 1 - N, invS);
        int y1 = min(max(y0 - RADIUS + j,     0), IMG_H - 1);
        int y2 = min(max(y0 - RADIUS + j + 1, 0), IMG_H - 1);
        v2f b;                                 // image[K=j][N], image[K=j+1][N]
        b.x = in[((size_t)y1 * IMG_W + (x0 + N)) * NCH + c];
        b.y = in[((size_t)y2 * IMG_W + (x0 + N)) * NCH + c];
        acc = __builtin_amdgcn_wmma_f32_16x16x4_f32(false, a, false, b,
                                                    (short)0, acc, false, false);
    }
#pragma unroll
    for (int g = 0; g < 8; ++g) {
        int y = y0 + g + 8 * half;
        int x = x0 + N;
        float bl = fminf(fmaxf(acc[g], 0.f), 1.f) * 255.0f;
        int dx = 1024 - x, dy = 1024 - y;
        int d2 = dx * dx + dy * dy;
        bool clean = (y >= 2 && y <= 2046 && x >= 2 && x <= 2046) && (d2 < SCLN2);
        size_t idx = ((size_t)y * IMG_W + x) * NCH + c;
        out[idx] = clean ? img[idx] : bl;
    }
}

// ---------------------------------------------------------------------------
extern "C" void kernel_launch(void* const* d_in, const int* in_sizes, int n_in,
                              void* d_out, int out_size, void* d_ws, size_t ws_size,
                              hipStream_t stream) {
    const float* img  = (const float*)d_in[0];
    const int*   offs = (const int*)d_in[1];
    float*       out  = (float*)d_out;

    unsigned long long* win = (unsigned long long*)d_ws;                 // 32 MB
    float* sw  = (float*)((char*)d_ws + (32ull << 20));                  // 48 MB
    float* tmp = (float*)((char*)d_ws + (80ull << 20));                  // 48 MB

    // Normalization: S = sum_i exp(-0.5*((i-8)/2)^2), float32 like the ref
    float S = 0.f;
    for (int i = 0; i < 17; ++i) {
        float tt = (float)(i - 8) / 2.0f;
        S += expf(-0.5f * tt * tt);
    }
    float invS = 1.0f / S;

    gb_init   <<<(NPIX + 255) / 256, 256, 0, stream>>>(win);
    gb_scatter<<<(NLOC + 255) / 256, 256, 0, stream>>>(offs, win);
    gb_gather <<<(NPIX + 255) / 256, 256, 0, stream>>>(img, win, sw);

    // 128x128 tiles x 3 channels = 49152 waves; 8 waves per 256-thread block
    const int nblk = (128 * 128 * NCH) / 8;   // 6144
    gb_blur_h<<<nblk, 256, 0, stream>>>(sw, tmp, invS);
    gb_blur_v<<<nblk, 256, 0, stream>>>(tmp, img, out, invS);
}